// Transformer_45664092291602
// MI455X (gfx1250) — compile-verified
//
#include <hip/hip_runtime.h>
#include <hip/hip_bf16.h>
#include <math.h>

#define N_NODES 10000
#define N_EDGES 320000
#define DIM     256
#define HEADS   4
#define DK      64
#define ND      (N_NODES * DIM)

typedef float v2f __attribute__((ext_vector_type(2)));
typedef float v8f __attribute__((ext_vector_type(8)));

// ---------------------------------------------------------------------------
// Degree accumulation: deg[src] += 1, deg[dst] += 1
// ---------------------------------------------------------------------------
__global__ __launch_bounds__(256) void deg_kernel(const int* __restrict__ src,
                                                  const int* __restrict__ dst,
                                                  float* __restrict__ deg, int e_cnt) {
    int e = blockIdx.x * 256 + threadIdx.x;
    if (e >= e_cnt) return;
    atomicAdd(&deg[src[e]], 1.0f);
    atomicAdd(&deg[dst[e]], 1.0f);
}

// ---------------------------------------------------------------------------
// Softmax over deg[0..n) in place (single block, 1024 threads)
// ---------------------------------------------------------------------------
__global__ __launch_bounds__(1024) void softmax_kernel(float* __restrict__ deg, int n) {
    __shared__ float red[1024];
    int t = threadIdx.x;
    float m = -1e30f;
    for (int i = t; i < n; i += 1024) m = fmaxf(m, deg[i]);
    red[t] = m;
    __syncthreads();
    for (int s = 512; s > 0; s >>= 1) {
        if (t < s) red[t] = fmaxf(red[t], red[t + s]);
        __syncthreads();
    }
    float mx = red[0];
    __syncthreads();
    float sum = 0.0f;
    for (int i = t; i < n; i += 1024) sum += expf(deg[i] - mx);
    red[t] = sum;
    __syncthreads();
    for (int s = 512; s > 0; s >>= 1) {
        if (t < s) red[t] += red[t + s];
        __syncthreads();
    }
    float inv = 1.0f / red[0];
    __syncthreads();
    for (int i = t; i < n; i += 1024) deg[i] = expf(deg[i] - mx) * inv;
}

// ---------------------------------------------------------------------------
// mV[i][j] = v[i][j] * center[i]
// ---------------------------------------------------------------------------
__global__ __launch_bounds__(256) void scale_kernel(const float* __restrict__ v,
                                                    const float* __restrict__ center,
                                                    float* __restrict__ mv, int total) {
    int idx = blockIdx.x * 256 + threadIdx.x;
    if (idx >= total) return;
    mv[idx] = v[idx] * center[idx >> 8];   // DIM == 256
}

// ---------------------------------------------------------------------------
// fp32 WMMA GEMM: C[M x Nc] = act(A[M x K] @ B[K x Nc] + bias)
// One wave32 computes a 16x64 output strip: 4 accumulators share one A
// fragment per K-step (V_WMMA_F32_16X16X4_F32). Nc must be a multiple of 64.
// act: 0 = none, 1 = relu6
// ---------------------------------------------------------------------------
__global__ __launch_bounds__(256) void gemm_wmma_f32(const float* __restrict__ A,
                                                     const float* __restrict__ B,
                                                     const float* __restrict__ bias,
                                                     float* __restrict__ C,
                                                     int M, int K, int Nc, int act) {
    int wave = (blockIdx.x * blockDim.x + threadIdx.x) >> 5;
    int lane = threadIdx.x & 31;
    int ntiles = Nc >> 6;                 // 64-wide strips
    int mt = wave / ntiles;
    int nt = wave - mt * ntiles;
    if (mt >= (M >> 4)) return;           // wave-uniform guard: EXEC stays full
    int m0 = mt << 4, n0 = nt << 6;

    int rc    = lane & 15;                // A row / B col within tile
    int khalf = (lane >> 4) << 1;         // 0 (lanes 0-15) or 2 (lanes 16-31)

    const float* Ap = A + (size_t)(m0 + rc) * K + khalf;
    const float* Bp = B + n0 + rc;

    v8f c0 = {}, c1 = {}, c2 = {}, c3 = {};
    #pragma unroll 2
    for (int kk = 0; kk < K; kk += 4) {
        v2f a;
        a.x = Ap[kk];
        a.y = Ap[kk + 1];
        const float* br0 = Bp + (size_t)(kk + khalf) * Nc;
        const float* br1 = br0 + Nc;
        v2f b0, b1, b2, b3;
        b0.x = br0[0];  b0.y = br1[0];
        b1.x = br0[16]; b1.y = br1[16];
        b2.x = br0[32]; b2.y = br1[32];
        b3.x = br0[48]; b3.y = br1[48];
        c0 = __builtin_amdgcn_wmma_f32_16x16x4_f32(false, a, false, b0, (short)0, c0, false, false);
        c1 = __builtin_amdgcn_wmma_f32_16x16x4_f32(false, a, false, b1, (short)0, c1, false, false);
        c2 = __builtin_amdgcn_wmma_f32_16x16x4_f32(false, a, false, b2, (short)0, c2, false, false);
        c3 = __builtin_amdgcn_wmma_f32_16x16x4_f32(false, a, false, b3, (short)0, c3, false, false);
    }

    int crow = m0 + ((lane >> 4) << 3);   // M = m0 + r + 8*(lane>=16)
    v8f acc[4] = {c0, c1, c2, c3};
    #pragma unroll
    for (int j = 0; j < 4; ++j) {
        int ccol = n0 + (j << 4) + rc;
        float bb = bias[ccol];
        #pragma unroll
        for (int r = 0; r < 8; ++r) {
            float val = acc[j][r] + bb;
            if (act == 1) val = fminf(fmaxf(val, 0.0f), 6.0f);
            C[(size_t)(crow + r) * Nc + ccol] = val;
        }
    }
}

// ---------------------------------------------------------------------------
// Edge attention: one thread per (edge, head)
//   s = exp(clip(dot(k[src], q[dst]) / 8, -5, 5))
//   wv[dst] += vv[src] * s ; z[dst] += s    (atomics)
// ---------------------------------------------------------------------------
__global__ __launch_bounds__(256) void edge_kernel(const int* __restrict__ src,
                                                   const int* __restrict__ dst,
                                                   const float* __restrict__ q,
                                                   const float* __restrict__ k,
                                                   const float* __restrict__ vv,
                                                   float* __restrict__ wv,
                                                   float* __restrict__ z, int e_cnt) {
    int tid = blockIdx.x * 256 + threadIdx.x;
    if (tid >= e_cnt * HEADS) return;
    int e = tid >> 2;
    int h = tid & 3;
    int sn = src[e], dn = dst[e];

    const float4* kp = (const float4*)(k + (size_t)sn * DIM + h * DK);
    const float4* qp = (const float4*)(q + (size_t)dn * DIM + h * DK);
    float score = 0.0f;
    #pragma unroll
    for (int i = 0; i < DK / 4; ++i) {
        float4 a = kp[i], b = qp[i];
        score += a.x * b.x + a.y * b.y + a.z * b.z + a.w * b.w;
    }
    float sc = fminf(fmaxf(score * 0.125f, -5.0f), 5.0f);  // /sqrt(64)
    float w = expf(sc);

    atomicAdd(&z[dn * HEADS + h], w);
    const float* vp = vv + (size_t)sn * DIM + h * DK;
    float* wp = wv + (size_t)dn * DIM + h * DK;
    #pragma unroll 4
    for (int i = 0; i < DK; ++i) atomicAdd(&wp[i], vp[i] * w);
}

// ---------------------------------------------------------------------------
// wv[n][h][d] /= z[n][h]   (in place)
// ---------------------------------------------------------------------------
__global__ __launch_bounds__(256) void div_kernel(float* __restrict__ wv,
                                                  const float* __restrict__ z, int total) {
    int idx = blockIdx.x * 256 + threadIdx.x;
    if (idx >= total) return;
    int n = idx >> 8;
    int h = (idx >> 6) & 3;
    wv[idx] = wv[idx] / z[n * HEADS + h];
}

// ---------------------------------------------------------------------------
// BatchNorm stats: one block per feature j; stats[j]=mean, stats[256+j]=var
// ---------------------------------------------------------------------------
__global__ __launch_bounds__(256) void bn_stats_kernel(const float* __restrict__ o,
                                                       float* __restrict__ stats, int n) {
    __shared__ float s1[256], s2[256];
    int j = blockIdx.x;
    float a = 0.0f, b = 0.0f;
    for (int i = threadIdx.x; i < n; i += 256) {
        float x = o[(size_t)i * DIM + j];
        a += x;
        b += x * x;
    }
    s1[threadIdx.x] = a;
    s2[threadIdx.x] = b;
    __syncthreads();
    for (int s = 128; s > 0; s >>= 1) {
        if (threadIdx.x < s) {
            s1[threadIdx.x] += s1[threadIdx.x + s];
            s2[threadIdx.x] += s2[threadIdx.x + s];
        }
        __syncthreads();
    }
    if (threadIdx.x == 0) {
        float mean = s1[0] / (float)n;
        stats[j]       = mean;
        stats[DIM + j] = s2[0] / (float)n - mean * mean;  // biased var
    }
}

// ---------------------------------------------------------------------------
// BN apply + ReLU6 in place: x = relu6(gamma*(o-mean)*rsqrt(var+eps)+beta)
// ---------------------------------------------------------------------------
__global__ __launch_bounds__(256) void bn_apply_kernel(float* __restrict__ o,
                                                       const float* __restrict__ stats,
                                                       const float* __restrict__ gamma,
                                                       const float* __restrict__ beta,
                                                       int total) {
    int idx = blockIdx.x * 256 + threadIdx.x;
    if (idx >= total) return;
    int j = idx & (DIM - 1);
    float mean = stats[j];
    float var  = stats[DIM + j];
    float x = gamma[j] * (o[idx] - mean) * rsqrtf(var + 1e-5f) + beta[j];
    o[idx] = fminf(fmaxf(x, 0.0f), 6.0f);
}

// ---------------------------------------------------------------------------
// out = hardswish(se) * x + v,  hardswish(s) = s * relu6(s+3)/6
// ---------------------------------------------------------------------------
__global__ __launch_bounds__(256) void final_kernel(const float* __restrict__ se,
                                                    const float* __restrict__ x,
                                                    const float* __restrict__ v,
                                                    float* __restrict__ out, int total) {
    int idx = blockIdx.x * 256 + threadIdx.x;
    if (idx >= total) return;
    float s  = se[idx];
    float hs = s * fminf(fmaxf(s + 3.0f, 0.0f), 6.0f) * (1.0f / 6.0f);
    out[idx] = hs * x[idx] + v[idx];
}

// ---------------------------------------------------------------------------
// Launch
// ---------------------------------------------------------------------------
extern "C" void kernel_launch(void* const* d_in, const int* in_sizes, int n_in,
                              void* d_out, int out_size, void* d_ws, size_t ws_size,
                              hipStream_t stream) {
    const float* v     = (const float*)d_in[0];
    const float* WQ    = (const float*)d_in[1];
    const float* bQ    = (const float*)d_in[2];
    const float* WK    = (const float*)d_in[3];
    const float* bK    = (const float*)d_in[4];
    const float* WV    = (const float*)d_in[5];
    const float* bV    = (const float*)d_in[6];
    const float* Wo    = (const float*)d_in[7];
    const float* bo    = (const float*)d_in[8];
    const float* W1    = (const float*)d_in[9];
    const float* b1    = (const float*)d_in[10];
    const float* W2    = (const float*)d_in[11];
    const float* b2    = (const float*)d_in[12];
    const float* gamma = (const float*)d_in[13];
    const float* beta  = (const float*)d_in[14];
    const int*   src   = (const int*)d_in[15];
    const int*   dst   = (const int*)d_in[16];
    float* out = (float*)d_out;

    // Workspace layout (floats)
    float* w      = (float*)d_ws;
    float* center = w;                        // N (deg -> softmax in place)
    float* bufA   = center + 10240;           // mV -> o -> x      [N*D]
    float* bufQ   = bufA + ND;                // q  -> h1          [N*D]
    float* bufK   = bufQ + ND;                // k  -> se          [N*D]
    float* bufV   = bufK + ND;                // vv                [N*D]
    float* bufW   = bufV + ND;                // wv (accum)        [N*D]
    float* zbuf   = bufW + ND;                // z                 [N*H]
    float* stats  = zbuf + 40960;             // mean/var          [512]

    // Zero accumulators each call (graph-capture safe)
    hipMemsetAsync(center, 0, N_NODES * sizeof(float), stream);
    hipMemsetAsync(bufW,   0, (size_t)ND * sizeof(float), stream);
    hipMemsetAsync(zbuf,   0, N_NODES * HEADS * sizeof(float), stream);

    const int eb = (N_EDGES + 255) / 256;
    const int nb = (ND + 255) / 256;

    // 1) degree + softmax center weighting
    deg_kernel<<<eb, 256, 0, stream>>>(src, dst, center, N_EDGES);
    softmax_kernel<<<1, 1024, 0, stream>>>(center, N_NODES);
    scale_kernel<<<nb, 256, 0, stream>>>(v, center, bufA, ND);

    // 2) Q/K/V projections: 625 row-tiles x 4 strips = 2500 waves -> 313 blocks
    const int gemm_blocks_256 = (625 * (DIM / 64) + 7) / 8;
    gemm_wmma_f32<<<gemm_blocks_256, 256, 0, stream>>>(bufA, WQ, bQ, bufQ, N_NODES, DIM, DIM, 0);
    gemm_wmma_f32<<<gemm_blocks_256, 256, 0, stream>>>(bufA, WK, bK, bufK, N_NODES, DIM, DIM, 0);
    gemm_wmma_f32<<<gemm_blocks_256, 256, 0, stream>>>(bufA, WV, bV, bufV, N_NODES, DIM, DIM, 0);

    // 3) edge attention scatter
    const int etb = (N_EDGES * HEADS + 255) / 256;
    edge_kernel<<<etb, 256, 0, stream>>>(src, dst, bufQ, bufK, bufV, bufW, zbuf, N_EDGES);
    div_kernel<<<nb, 256, 0, stream>>>(bufW, zbuf, ND);

    // 4) output projection (wv/z) @ Wo + bo -> o (reuse bufA)
    gemm_wmma_f32<<<gemm_blocks_256, 256, 0, stream>>>(bufW, Wo, bo, bufA, N_NODES, DIM, DIM, 0);

    // 5) BatchNorm (batch stats) + ReLU6 in place
    bn_stats_kernel<<<DIM, 256, 0, stream>>>(bufA, stats, N_NODES);
    bn_apply_kernel<<<nb, 256, 0, stream>>>(bufA, stats, gamma, beta, ND);

    // 6) SE block: x @ W1 -> relu6 -> @ W2
    const int gemm_blocks_64 = (625 * (64 / 64) + 7) / 8;   // 625 waves -> 79 blocks
    gemm_wmma_f32<<<gemm_blocks_64, 256, 0, stream>>>(bufA, W1, b1, bufQ, N_NODES, DIM, 64, 1);
    gemm_wmma_f32<<<gemm_blocks_256, 256, 0, stream>>>(bufQ, W2, b2, bufK, N_NODES, 64, DIM, 0);

    // 7) hardswish(se) * x + v
    final_kernel<<<nb, 256, 0, stream>>>(bufK, bufA, v, out, ND);
}